// Gumbel_Generator_nc_18159121727740
// MI455X (gfx1250) — compile-verified
//
#include <hip/hip_runtime.h>
#include <cstdint>
#include <cstddef>

#define SZ   8192
#define DN   2048
#define S_   6144                 // SZ - DN
#define BASE2 12582912LL          // S_ * DN
#define TS   32

typedef float v2f __attribute__((ext_vector_type(2)));
typedef float v4f __attribute__((ext_vector_type(4)));

// y0 = softmax over {a0,a1}[0] = 1 / (1 + exp(a1 - a0)),  a_i = (g_i + gumbel(u_i))/10
__device__ __forceinline__ float sig_gumbel(float g0, float g1, float u0, float u1) {
    const float EPS = 1e-20f;
    float gum0 = -__logf(-__logf(u0 + EPS) + EPS);
    float gum1 = -__logf(-__logf(u1 + EPS) + EPS);
    float d = ((g1 + gum1) - (g0 + gum0)) * 0.1f;
    return __builtin_amdgcn_rcpf(1.0f + __expf(d));
}

// CDNA5 async global -> LDS (ASYNCcnt-tracked DMA into LDS, no VGPR staging).
// Inputs are read exactly once -> non-temporal load hint.
__device__ __forceinline__ void async_ld_b128(unsigned lds_byte_off, const void* gaddr) {
    asm volatile("global_load_async_to_lds_b128 %0, %1, off th:TH_LOAD_NT"
                 :: "v"(lds_byte_off), "v"(gaddr) : "memory");
}
__device__ __forceinline__ void wait_asynccnt0() {
    asm volatile("s_wait_asynccnt 0" ::: "memory");
}

// ---------------- kernel 1: zero the top-left 6144x6144 block -------------
// One block per row; 6 x b128 NT stores per thread (streaming, don't pollute L2).
__global__ __launch_bounds__(256) void zero_kernel(float* __restrict__ out) {
    v4f* rp = (v4f*)(out + (size_t)blockIdx.x * SZ);   // row 0..6143
    v4f z = (v4f)(0.0f);
#pragma unroll
    for (int j = 0; j < 6; ++j)
        __builtin_nontemporal_store(z, rp + j * 256 + threadIdx.x);
}

// ------- kernel 2: dense block A (rows [0,s), cols [s,SZ)) + mirror -------
// e = (r0+row)*DN + k0 + k ; 32x32 tile per workgroup, block = (32,8)
__global__ __launch_bounds__(256) void blockA_kernel(const v2f* __restrict__ gen,
                                                     const v2f* __restrict__ uu,
                                                     float* __restrict__ out) {
    __shared__ v2f   gt[TS * TS];         // 8 KB raw gen tile
    __shared__ v2f   ut[TS * TS];         // 8 KB raw u tile
    __shared__ float vt[TS][TS + 1];      // computed values, padded (bank-conflict free)

    const int k0  = blockIdx.x * TS;      // 0..2047
    const int r0  = blockIdx.y * TS;      // 0..6143
    const int tid = threadIdx.y * 32 + threadIdx.x;

    // Stage inputs via async DMA: 32 rows * 32 v2f = 512 x 16B chunks per tile.
    // Base element index (r0+row)*2048 + k0 is even -> 16B-aligned B128.
#pragma unroll
    for (int p = 0; p < 2; ++p) {
        int c   = p * 256 + tid;          // 0..511
        int row = c >> 4;                 // 16 chunks per row
        int ch  = c & 15;
        size_t ebase = (size_t)(r0 + row) * DN + (size_t)k0 + (size_t)(ch * 2);
        async_ld_b128((unsigned)(uintptr_t)(const void*)&gt[row * TS + ch * 2], gen + ebase);
        async_ld_b128((unsigned)(uintptr_t)(const void*)&ut[row * TS + ch * 2], uu  + ebase);
    }
    wait_asynccnt0();
    __syncthreads();

    const int lj = threadIdx.x;
#pragma unroll
    for (int rr = threadIdx.y; rr < TS; rr += 8) {
        v2f g = gt[rr * TS + lj];
        v2f w = ut[rr * TS + lj];
        vt[rr][lj] = sig_gumbel(g.x, g.y, w.x, w.y);
    }
    __syncthreads();

#pragma unroll
    for (int rr = threadIdx.y; rr < TS; rr += 8) {
        // block A: row-major, coalesced, streaming store
        __builtin_nontemporal_store(vt[rr][lj],
            out + (size_t)(r0 + rr) * SZ + (size_t)(S_ + k0 + lj));
        // block A^T: coalesced streaming store, transposed LDS read (pitch 33)
        __builtin_nontemporal_store(vt[lj][rr],
            out + (size_t)(S_ + k0 + rr) * SZ + (size_t)(r0 + lj));
    }
}

// -- kernel 3: bottom-right 2048x2048 strict upper triangle + mirror + diag 0 --
// e = BASE2 + i*(4095-i)/2 + (j-i-1), i<j. 32x32 tiles; block = (32,8)
__global__ __launch_bounds__(256) void triu_kernel(const v2f* __restrict__ gen,
                                                   const v2f* __restrict__ uu,
                                                   float* __restrict__ out) {
    __shared__ float vt[TS][TS + 1];
    const int btj = blockIdx.x, bti = blockIdx.y;
    if (bti > btj) return;                       // lower-triangular tiles handled by mirror
    const int ti0 = bti * TS, tj0 = btj * TS;
    const int lj  = threadIdx.x;

    if (bti != btj) {                            // strictly upper tile: full 32x32 valid
#pragma unroll
        for (int rr = threadIdx.y; rr < TS; rr += 8) {
            int i = ti0 + rr;
            long long e = BASE2 + (long long)i * (2 * DN - 1 - i) / 2
                        + (long long)(tj0 + lj - i - 1);
            v2f g = __builtin_nontemporal_load(gen + e);
            v2f w = __builtin_nontemporal_load(uu + e);
            vt[rr][lj] = sig_gumbel(g.x, g.y, w.x, w.y);
        }
        __syncthreads();
#pragma unroll
        for (int rr = threadIdx.y; rr < TS; rr += 8) {
            __builtin_nontemporal_store(vt[rr][lj],
                out + (size_t)(S_ + ti0 + rr) * SZ + (size_t)(S_ + tj0 + lj));
            __builtin_nontemporal_store(vt[lj][rr],
                out + (size_t)(S_ + tj0 + rr) * SZ + (size_t)(S_ + ti0 + lj));
        }
    } else {                                     // diagonal tile: i<j computed, mirrored in-tile
#pragma unroll
        for (int rr = threadIdx.y; rr < TS; rr += 8) {
            int i = ti0 + rr;
            float v = 0.0f;
            if (lj > rr) {
                long long e = BASE2 + (long long)i * (2 * DN - 1 - i) / 2
                            + (long long)(tj0 + lj - i - 1);
                v2f g = __builtin_nontemporal_load(gen + e);
                v2f w = __builtin_nontemporal_load(uu + e);
                v = sig_gumbel(g.x, g.y, w.x, w.y);
            }
            vt[rr][lj] = v;                      // diagonal & below store 0 / get mirrored
        }
        __syncthreads();
#pragma unroll
        for (int rr = threadIdx.y; rr < TS; rr += 8) {
            float v = (lj >= rr) ? vt[rr][lj] : vt[lj][rr];
            __builtin_nontemporal_store(v,
                out + (size_t)(S_ + ti0 + rr) * SZ + (size_t)(S_ + tj0 + lj));
        }
    }
}

extern "C" void kernel_launch(void* const* d_in, const int* in_sizes, int n_in,
                              void* d_out, int out_size, void* d_ws, size_t ws_size,
                              hipStream_t stream) {
    (void)in_sizes; (void)n_in; (void)out_size; (void)d_ws; (void)ws_size;
    const v2f* gen = (const v2f*)d_in[0];   // (E,2) fp32
    const v2f* uu  = (const v2f*)d_in[1];   // (E,2) fp32
    float* out = (float*)d_out;             // 8192*8192 fp32

    zero_kernel  <<<dim3(6144),    dim3(256),   0, stream>>>(out);
    blockA_kernel<<<dim3(64, 192), dim3(32, 8), 0, stream>>>(gen, uu, out);
    triu_kernel  <<<dim3(64, 64),  dim3(32, 8), 0, stream>>>(gen, uu, out);
}